// ConvexHullModel_36129264894520
// MI455X (gfx1250) — compile-verified
//
#include <hip/hip_runtime.h>
#include <hip/hip_bf16.h>

typedef __attribute__((ext_vector_type(2))) float v2f;
typedef __attribute__((ext_vector_type(8))) float v8f;

__device__ __forceinline__ void atomic_add_f32(float* p, float v) {
  unsafeAtomicAdd(p, v);   // -> GLOBAL_ATOMIC_ADD_F32 (no CAS loop)
}

// ---------------------------------------------------------------- utilities
__global__ __launch_bounds__(256) void k_fill(float* __restrict__ p, float v, int n) {
  int i = blockIdx.x * blockDim.x + threadIdx.x;
  if (i < n) p[i] = v;
}

// deg[dst] += 1 for every real edge (deg pre-initialized to 1.0 for self-loop)
__global__ __launch_bounds__(256) void k_deg(const int* __restrict__ ei,
                                             float* __restrict__ deg, int E) {
  int e = blockIdx.x * blockDim.x + threadIdx.x;
  if (e < E) atomic_add_f32(&deg[ei[E + e]], 1.0f);
}

// in-place deg -> dinv = rsqrt(deg)
__global__ __launch_bounds__(256) void k_rsqrt(float* __restrict__ d, int n) {
  int i = blockIdx.x * blockDim.x + threadIdx.x;
  if (i < n) { float v = d[i]; d[i] = (v > 0.f) ? rsqrtf(v) : 0.f; }
}

// ---------------------------------------------------------------- WMMA matmul
// h = relu?(in + bprev) @ W via V_WMMA_F32_16X16X4_F32, 16 nodes per wave.
// Also fuses the self-loop init: agg[m] = h[m] * dinv[m]^2 (zero-initializes agg).
// Safe for in == agg: each wave reads only its own 16 nodes before writing them.
__global__ __launch_bounds__(256) void k_mm_wmma(const float* __restrict__ in,
                                                 const float* __restrict__ W,
                                                 const float* __restrict__ bprev,
                                                 int fuse_prev,
                                                 const float* __restrict__ dinv,
                                                 float* __restrict__ out,
                                                 float* __restrict__ agg, int n) {
  const int lane = threadIdx.x & 31;
  const int wave = blockIdx.x * (blockDim.x >> 5) + (threadIdx.x >> 5);
  const int m0 = wave << 4;                 // 16 nodes per wave
  if (m0 >= n) return;                      // wave-uniform: EXEC all-1s at WMMA

  const int r = lane & 15;

  // A: 16x4 (nodes x feat, K=2,3 zero-padded). lanes<16: K=0,1; lanes>=16: K=2,3.
  v2f a; a.x = 0.f; a.y = 0.f;
  if (lane < 16) {
    const int node = m0 + r;
    float x0 = 0.f, x1 = 0.f;
    if (node < n) { x0 = in[node * 2 + 0]; x1 = in[node * 2 + 1]; }
    if (fuse_prev) { x0 = fmaxf(x0 + bprev[0], 0.f); x1 = fmaxf(x1 + bprev[1], 0.f); }
    a.x = x0; a.y = x1;
  }
  // B: 4x16, rows K=2,3 and cols N=2..15 zero. W row-major 2x2: W[k][j]=W[k*2+j].
  v2f b; b.x = 0.f; b.y = 0.f;
  if (lane < 16 && r < 2) { b.x = W[0 * 2 + r]; b.y = W[1 * 2 + r]; }

  v8f c = {};
  c = __builtin_amdgcn_wmma_f32_16x16x4_f32(false, a, false, b, (short)0, c,
                                            false, false);

  // D: VGPR j -> M=j (lanes 0-15, N=lane) / M=8+j (lanes 16-31, N=lane-16).
  // Pull column-1 value next to column-0 value so lanes 0 and 16 can do b64 stores.
  float oth[8];
#pragma unroll
  for (int j = 0; j < 8; ++j) oth[j] = __shfl_xor(c[j], 1, 32);

  if ((lane & 15) == 0) {                   // lanes 0 and 16
    const int mb = m0 + ((lane >> 4) << 3); // +8 for upper half-wave
    if (mb + 8 <= n) {
#pragma unroll
      for (int j = 0; j < 8; ++j) {
        const int m = mb + j;
        const float h0 = c[j], h1 = oth[j];
        *(float2*)(out + (size_t)m * 2) = make_float2(h0, h1);
        float w = dinv[m]; w *= w;
        *(float2*)(agg + (size_t)m * 2) = make_float2(h0 * w, h1 * w);
      }
    } else {
      for (int j = 0; j < 8; ++j) {
        const int m = mb + j;
        if (m < n) {
          const float h0 = c[j], h1 = oth[j];
          *(float2*)(out + (size_t)m * 2) = make_float2(h0, h1);
          float w = dinv[m]; w *= w;
          *(float2*)(agg + (size_t)m * 2) = make_float2(h0 * w, h1 * w);
        }
      }
    }
  }
}

// ---------------------------------------------------------------- aggregation
// agg[dst] += h[src] * dinv[src] * dinv[dst]  (one thread per edge)
__global__ __launch_bounds__(256) void k_scatter(const float* __restrict__ h,
                                                 const float* __restrict__ dinv,
                                                 const int* __restrict__ ei,
                                                 float* __restrict__ agg, int E) {
  int e = blockIdx.x * blockDim.x + threadIdx.x;
  if (e >= E) return;
  int s = ei[e];
  int d = ei[E + e];
  float w = dinv[s] * dinv[d];
  float2 hv = *(const float2*)(h + (size_t)s * 2);
  atomic_add_f32(&agg[d * 2 + 0], hv.x * w);
  atomic_add_f32(&agg[d * 2 + 1], hv.y * w);
}

// pooled[batch[i]] += relu(agg[i] + b3)
__global__ __launch_bounds__(256) void k_pool(const float* __restrict__ agg,
                                              const float* __restrict__ b3,
                                              const int* __restrict__ batch,
                                              float* __restrict__ pooled, int n) {
  int i = blockIdx.x * blockDim.x + threadIdx.x;
  if (i >= n) return;
  float v0 = fmaxf(agg[i * 2 + 0] + b3[0], 0.f);
  float v1 = fmaxf(agg[i * 2 + 1] + b3[1], 0.f);
  int g = batch[i];
  atomic_add_f32(&pooled[g * 2 + 0], v0);
  atomic_add_f32(&pooled[g * 2 + 1], v1);
}

// out[g] = pooled[g] @ Wr + br
__global__ __launch_bounds__(256) void k_readout(const float* __restrict__ pooled,
                                                 const float* __restrict__ Wr,
                                                 const float* __restrict__ br,
                                                 float* __restrict__ out, int G) {
  int g = blockIdx.x * blockDim.x + threadIdx.x;
  if (g < G) out[g] = pooled[g * 2 + 0] * Wr[0] + pooled[g * 2 + 1] * Wr[1] + br[0];
}

// ---------------------------------------------------------------- launch
extern "C" void kernel_launch(void* const* d_in, const int* in_sizes, int n_in,
                              void* d_out, int out_size, void* d_ws, size_t ws_size,
                              hipStream_t stream) {
  (void)n_in; (void)ws_size;
  const float* x     = (const float*)d_in[0];
  const int*   ei    = (const int*)d_in[1];
  const int*   batch = (const int*)d_in[2];
  const float* W1 = (const float*)d_in[3];
  const float* b1 = (const float*)d_in[4];
  const float* W2 = (const float*)d_in[5];
  const float* b2 = (const float*)d_in[6];
  const float* W3 = (const float*)d_in[7];
  const float* b3 = (const float*)d_in[8];
  const float* Wr = (const float*)d_in[9];
  const float* br = (const float*)d_in[10];

  const int N = in_sizes[2];
  const int E = in_sizes[1] / 2;
  const int G = out_size;

  float* ws     = (float*)d_ws;
  float* dinv   = ws;                       // N   (deg, then rsqrt in place)
  float* bufH   = ws + (size_t)N;           // 2N  (h = X @ W)
  float* bufAgg = ws + (size_t)3 * N;       // 2N  (aggregated output)
  float* pooled = ws + (size_t)5 * N;       // 2G

  const int B = 256;
  auto gr = [&](int n) { return (n + B - 1) / B; };
  const int mmBlocks = (((N + 15) / 16) * 32 + B - 1) / B;

  // symmetric normalization
  k_fill<<<gr(N), B, 0, stream>>>(dinv, 1.0f, N);          // self-loop degree
  k_fill<<<gr(2 * G), B, 0, stream>>>(pooled, 0.0f, 2 * G);
  k_deg<<<gr(E), B, 0, stream>>>(ei, dinv, E);
  k_rsqrt<<<gr(N), B, 0, stream>>>(dinv, N);

  // layer 1: h = x @ W1            (agg initialized with self-loop term)
  k_mm_wmma<<<mmBlocks, B, 0, stream>>>(x, W1, b1, 0, dinv, bufH, bufAgg, N);
  k_scatter<<<gr(E), B, 0, stream>>>(bufH, dinv, ei, bufAgg, E);

  // layer 2: h = relu(agg + b1) @ W2   (in-place read/overwrite of bufAgg is safe)
  k_mm_wmma<<<mmBlocks, B, 0, stream>>>(bufAgg, W2, b1, 1, dinv, bufH, bufAgg, N);
  k_scatter<<<gr(E), B, 0, stream>>>(bufH, dinv, ei, bufAgg, E);

  // layer 3: h = relu(agg + b2) @ W3
  k_mm_wmma<<<mmBlocks, B, 0, stream>>>(bufAgg, W3, b2, 1, dinv, bufH, bufAgg, N);
  k_scatter<<<gr(E), B, 0, stream>>>(bufH, dinv, ei, bufAgg, E);

  // pool (fused bias+relu) + readout
  k_pool<<<gr(N), B, 0, stream>>>(bufAgg, b3, batch, pooled, N);
  k_readout<<<gr(G), B, 0, stream>>>(pooled, Wr, br, (float*)d_out, G);
}